// EditDistance_54795192763001
// MI455X (gfx1250) — compile-verified
//
#include <hip/hip_runtime.h>

#define BATCH 8192
#define LSEQ  64
#define TPB   128            // pairs per block (4 waves of 32)
#define APAD  65             // padded int stride for LDS A-tile (bank-conflict-free)

// ---- CDNA5 async global->LDS availability probe (compile-time) ----
#ifdef __has_builtin
#  if __has_builtin(__builtin_amdgcn_global_load_async_to_lds_b128)
#    define HAVE_ASYNC_LDS 1
#  endif
#  if __has_builtin(__builtin_amdgcn_s_wait_asynccnt)
#    define HAVE_WAIT_ASYNC_BUILTIN 1
#  endif
#endif
#ifndef HAVE_ASYNC_LDS
#  define HAVE_ASYNC_LDS 0
#endif
#ifndef HAVE_WAIT_ASYNC_BUILTIN
#  define HAVE_WAIT_ASYNC_BUILTIN 0
#endif

// 16-byte int vector type matching the builtin's parameter type
typedef int v4i __attribute__((vector_size(16)));
typedef __attribute__((address_space(1))) v4i g_v4i;   // global
typedef __attribute__((address_space(3))) v4i l_v4i;   // LDS

__global__ __launch_bounds__(TPB) void
EditDistance_kernel(const int* __restrict__ tokens1,
                    const int* __restrict__ tokens2,
                    const int* __restrict__ len1,
                    const int* __restrict__ len2,
                    const float* __restrict__ sub_cost,
                    float* __restrict__ out)
{
    __shared__ int lds_a[TPB * APAD];           // 128 * 65 * 4 = 33,280 B

    const int tid = threadIdx.x;
    const int p0  = blockIdx.x * TPB;
    const int p   = p0 + tid;

    // ---- Stage the A-token tile (TPB rows x 64 ints) into LDS ----
    // Chunk f = k*TPB + tid covers row r = f/16, 16B-quad q = f%16.
    // Lanes of a wave touch contiguous 512B of global memory per k.
#pragma unroll
    for (int k = 0; k < 16; ++k) {
        const int f = k * TPB + tid;
        const int r = f >> 4;
        const int q = f & 15;
        const int* gsrc = tokens1 + (size_t)(p0 + r) * LSEQ + q * 4;
        int*       ldst = &lds_a[r * APAD + q * 4];
#if HAVE_ASYNC_LDS
        // CDNA5: GLOBAL_LOAD_ASYNC_TO_LDS_B128 (tracked by ASYNCcnt)
        __builtin_amdgcn_global_load_async_to_lds_b128(
            (g_v4i*)gsrc, (l_v4i*)ldst, /*offset=*/0, /*cpol=*/0);
#else
        const int4 v = *(const int4*)gsrc;
        *(int4*)ldst = v;
#endif
    }
#if HAVE_ASYNC_LDS
#  if HAVE_WAIT_ASYNC_BUILTIN
    __builtin_amdgcn_s_wait_asynccnt(0);
#  else
    asm volatile("s_wait_asynccnt 0" ::: "memory");
#  endif
#endif
    __syncthreads();

    // ---- Per-thread state ----
    const int   la = len1[p];
    const int   lb = len2[p];
    const float c  = sub_cost[0];

    // B tokens into registers (static indices after full unroll)
    int breg[LSEQ];
    {
        const int4* bq = (const int4*)(tokens2 + (size_t)p * LSEQ);
#pragma unroll
        for (int k = 0; k < 16; ++k) {
            const int4 v = bq[k];
            breg[4 * k + 0] = v.x;
            breg[4 * k + 1] = v.y;
            breg[4 * k + 2] = v.z;
            breg[4 * k + 3] = v.w;
        }
    }

    // DP row D[i-1][0..64] kept entirely in VGPRs (inner loop fully unrolled)
    float row[LSEQ + 1];
#pragma unroll
    for (int j = 0; j <= LSEQ; ++j) row[j] = (float)j;

    float ans = (float)lb;                      // covers la == 0 (D[0][lb] = lb)

    const int abase = tid * APAD;
#pragma unroll 1
    for (int i = 1; i <= LSEQ; ++i) {
        const int   ai   = lds_a[abase + (i - 1)];   // ds_load_b32, conflict-free
        const bool  hit  = (i == la);
        float       left = (float)i;                  // D[i][0] = i
        float       diag = row[0];                    // D[i-1][0]
        row[0] = left;
        if (hit && lb == 0) ans = left;
#pragma unroll
        for (int j = 1; j <= LSEQ; ++j) {
            const float up   = row[j];                // D[i-1][j]
            const float cost = (ai == breg[j - 1]) ? 0.0f : c;
            const float v    = fminf(fminf(up + 1.0f, left + 1.0f), diag + cost);
            row[j] = v;
            if (hit && j == lb) ans = v;              // capture D[la][lb]
            diag = up;
            left = v;
        }
    }

    out[p] = ans;
}

extern "C" void kernel_launch(void* const* d_in, const int* in_sizes, int n_in,
                              void* d_out, int out_size, void* d_ws, size_t ws_size,
                              hipStream_t stream)
{
    const int*   tokens1 = (const int*)d_in[0];
    const int*   tokens2 = (const int*)d_in[1];
    const int*   len1    = (const int*)d_in[2];
    const int*   len2    = (const int*)d_in[3];
    const float* sub     = (const float*)d_in[4];
    float*       out     = (float*)d_out;

    dim3 grid(BATCH / TPB);
    dim3 block(TPB);
    EditDistance_kernel<<<grid, block, 0, stream>>>(tokens1, tokens2, len1, len2, sub, out);
}